// StoppingTimeProximityLoss_64372969832510
// MI455X (gfx1250) — compile-verified
//
#include <hip/hip_runtime.h>

typedef __attribute__((ext_vector_type(2))) float v2f;
typedef __attribute__((ext_vector_type(8))) float v8f;

#define N_ROWS 4096
#define T_LEN  365
#define C_CLS  32
#define T_PAD  368            // 23 * 16, padded column count for WMMA tiles
#define NB     128            // pass-1 blocks
#define WPB    8              // waves per block (256 threads, wave32)
#define ROWS_PER_WAVE (N_ROWS / (NB * WPB))   // = 4

// workspace layout (float offsets)
#define WS_CE   0                      // NB * T_PAD
#define WS_EARL (NB * T_PAD)           // NB
#define WS_A    (WS_EARL + NB)         // NB * C_CLS
#define WS_B    (WS_A + NB * C_CLS)    // NB * C_CLS
#define WS_D    (WS_B + NB * C_CLS)    // NB * C_CLS

__global__ __launch_bounds__(256) void stpl_pass1(
    const float* __restrict__ lcp,     // (N,T,C) log-probs
    const float* __restrict__ tsl,     // (N,T) timestamps_left
    const int*   __restrict__ ytrue,   // (N,T) labels (constant per row)
    float* __restrict__ ws)
{
    __shared__ float lds_ce[WPB][T_PAD];
    __shared__ float lds_earl[WPB];
    __shared__ float lds_A[WPB][C_CLS];
    __shared__ float lds_B[WPB][C_CLS];
    __shared__ float lds_D[WPB][C_CLS];

    const int tid  = threadIdx.x;
    const int w    = tid >> 5;
    const int lane = tid & 31;
    const int b    = blockIdx.x;

    for (int i = tid; i < WPB * C_CLS; i += 256) {
        (&lds_A[0][0])[i] = 0.f;
        (&lds_B[0][0])[i] = 0.f;
        (&lds_D[0][0])[i] = 0.f;
    }
    __syncthreads();

    const float invT = 1.0f / (float)T_LEN;

    float ceacc[12];
#pragma unroll
    for (int j = 0; j < 12; ++j) ceacc[j] = 0.f;
    float earl = 0.f;

    const int gw = b * WPB + w;          // global wave id, 0..1023
    for (int r = 0; r < ROWS_PER_WAVE; ++r) {
        const int n = gw * ROWS_PER_WAVE + r;
        const int y = ytrue[(size_t)n * T_LEN];
        const float* rowbase = lcp + ((size_t)n * T_LEN * C_CLS) + y;
        const float* tslrow  = tsl + (size_t)n * T_LEN;

        int   tmin  = T_LEN;   // sentinel: no zero seen
        float lpsel = 0.f;

        // 11 unguarded iterations: t = lane + 32*j <= 351 < 365 (max MLP)
#pragma unroll
        for (int j = 0; j < 11; ++j) {
            const int t = lane + 32 * j;
            const float lp = rowbase[(size_t)t * C_CLS];
            const float tl = tslrow[t];
            ceacc[j] += lp;
            const float p = __expf(lp);
            earl += p * (1.f - (float)t * invT) * (1.f - tl * invT);
            if (tl == 0.f && t < tmin) { tmin = t; lpsel = lp; }
        }
        {   // guarded tail, j = 11 (t = lane + 352)
            const int t = lane + 352;
            if (t < T_LEN) {
                const float lp = rowbase[(size_t)t * C_CLS];
                const float tl = tslrow[t];
                ceacc[11] += lp;
                const float p = __expf(lp);
                earl += p * (1.f - (float)t * invT) * (1.f - tl * invT);
                if (tl == 0.f && t < tmin) { tmin = t; lpsel = lp; }
            }
        }

        // wave32 min-with-payload reduce -> (t_final, lp at t_final)
#pragma unroll
        for (int off = 16; off >= 1; off >>= 1) {
            const int   ot  = __shfl_down(tmin,  off, 32);
            const float olp = __shfl_down(lpsel, off, 32);
            if (ot < tmin) { tmin = ot; lpsel = olp; }
        }
        tmin  = __shfl(tmin,  0, 32);
        lpsel = __shfl(lpsel, 0, 32);

        if (lane == 0) {   // per-wave class moments (no cross-wave races)
            const float e  = __expf(lpsel);
            const float tf = (float)tmin;
            lds_A[w][y] += e;
            lds_B[w][y] += e * tf;
            lds_D[w][y] += e * tf * tf;
        }
    }

    // per-wave CE partials to LDS (padding cols 365..367 get 0 via unused accs)
#pragma unroll
    for (int j = 0; j < 12; ++j) {
        const int t = lane + 32 * j;
        if (t < T_PAD) lds_ce[w][t] = ceacc[j];
    }
    // earliness wave reduce
#pragma unroll
    for (int off = 16; off >= 1; off >>= 1)
        earl += __shfl_down(earl, off, 32);
    if (lane == 0) lds_earl[w] = earl;
    __syncthreads();

    // fixed-order block combine -> workspace (deterministic)
    for (int t = tid; t < T_PAD; t += 256) {
        float s = 0.f;
#pragma unroll
        for (int q = 0; q < WPB; ++q) s += lds_ce[q][t];
        ws[WS_CE + b * T_PAD + t] = s;
    }
    if (tid == 0) {
        float s = 0.f;
        for (int q = 0; q < WPB; ++q) s += lds_earl[q];
        ws[WS_EARL + b] = s;
    }
    if (tid < C_CLS) {
        float sa = 0.f, sb = 0.f, sd = 0.f;
        for (int q = 0; q < WPB; ++q) {
            sa += lds_A[q][tid]; sb += lds_B[q][tid]; sd += lds_D[q][tid];
        }
        ws[WS_A + b * C_CLS + tid] = sa;
        ws[WS_B + b * C_CLS + tid] = sb;
        ws[WS_D + b * C_CLS + tid] = sd;
    }
}

__global__ __launch_bounds__(256) void stpl_pass2(
    const float* __restrict__ ws, float* __restrict__ out)
{
    __shared__ float red_ce[T_PAD];
    __shared__ float cls_prox[C_CLS];
    __shared__ float sc[2];   // [0]=earliness total, [1]=proximity numerator

    const int tid  = threadIdx.x;
    const int w    = tid >> 5;
    const int lane = tid & 31;
    const int l    = lane & 15;
    const int half = lane >> 4;

    const float* ce = ws + WS_CE;   // [NB][T_PAD]

    // A = 16x4 all-ones (f32): lanes 0-15 rows M, VGPR0=K0/K2, VGPR1=K1/K3
    v2f am; am.x = 1.0f; am.y = 1.0f;

    // WMMA reduction of CE partials over NB block-rows, per 16-col t-tile.
    // Wave-uniform loop bounds -> EXEC all-ones at the WMMA site.
    for (int tt = w; tt < T_PAD / 16; tt += WPB) {   // 23 tiles over 8 waves
        const int tbase = tt * 16;
        v8f c = {};
        for (int kb = 0; kb < NB / 4; ++kb) {
            // B 4x16 layout: VGPR0 = K {0|2}, VGPR1 = K {1|3} (lanes 0-15 | 16-31)
            v2f bm;
            bm.x = ce[(size_t)(4 * kb + 2 * half)     * T_PAD + tbase + l];
            bm.y = ce[(size_t)(4 * kb + 2 * half + 1) * T_PAD + tbase + l];
            c = __builtin_amdgcn_wmma_f32_16x16x4_f32(
                    /*neg_a=*/false, am, /*neg_b=*/false, bm,
                    /*c_mod=*/(short)0, c, /*reuse_a=*/false, /*reuse_b=*/false);
        }
        // D rows identical (ones-A); row M=0 lives in lanes 0-15 of c[0]
        if (lane < 16) red_ce[tbase + lane] = c[0];
    }
    __syncthreads();

    if (tid < C_CLS) {   // per-class moments, fixed order over blocks
        float sa = 0.f, sb = 0.f, sd = 0.f;
        for (int q = 0; q < NB; ++q) {
            sa += ws[WS_A + q * C_CLS + tid];
            sb += ws[WS_B + q * C_CLS + tid];
            sd += ws[WS_D + q * C_CLS + tid];
        }
        cls_prox[tid] = sa * sd - sb * sb;
    }
    __syncthreads();
    if (tid == 0) {
        float e = 0.f;
        for (int q = 0; q < NB; ++q) e += ws[WS_EARL + q];
        float p = 0.f;
        for (int cc = 0; cc < C_CLS; ++cc) p += cls_prox[cc];
        sc[0] = e; sc[1] = p;
    }
    __syncthreads();

    const float Tf = (float)T_LEN;
    const float a0 = 1.0f / 3.0f, a1 = 1.0f / 3.0f, a2 = 1.0f / 3.0f;
    const float earl_mean = sc[0] / (float)N_ROWS;
    const float prox      = sc[1] / (Tf * Tf);
    for (int t = tid; t < T_LEN; t += 256) {
        const float cls = -red_ce[t] / (float)N_ROWS;   // mean_n cross-entropy
        out[t] = a0 * cls - a1 * earl_mean - a2 * prox;
    }
}

extern "C" void kernel_launch(void* const* d_in, const int* in_sizes, int n_in,
                              void* d_out, int out_size, void* d_ws, size_t ws_size,
                              hipStream_t stream) {
    const float* lcp   = (const float*)d_in[0];  // log_class_probabilities (N,T,C)
    const float* tsl   = (const float*)d_in[1];  // timestamps_left (N,T)
    const int*   ytrue = (const int*)d_in[2];    // y_true (N,T)
    float* out = (float*)d_out;                  // (T,)
    float* ws  = (float*)d_ws;                   // needs ~238 KB

    hipLaunchKernelGGL(stpl_pass1, dim3(NB), dim3(256), 0, stream, lcp, tsl, ytrue, ws);
    hipLaunchKernelGGL(stpl_pass2, dim3(1),  dim3(256), 0, stream, ws, out);
}